// KappaGCN_11647951307483
// MI455X (gfx1250) — compile-verified
//
#include <hip/hip_runtime.h>
#include <hip/hip_bf16.h>
#include <math.h>

typedef float v2f __attribute__((ext_vector_type(2)));
typedef float v8f __attribute__((ext_vector_type(8)));

#define DD 32
#define CC 16
#define GCOLS 48
#define EPSF 1e-15f

__device__ __forceinline__ float wave_sum(float v) {
  #pragma unroll
  for (int off = 16; off > 0; off >>= 1)
    v += __shfl_xor(v, off, 32);
  return v;
}

__device__ __forceinline__ float clip1(float x) {
  return fminf(x, 1.0f - 1e-7f);
}

__device__ __forceinline__ v8f wmma4(v2f a, v2f b, v8f c) {
  return __builtin_amdgcn_wmma_f32_16x16x4_f32(
      false, a, false, b, (short)0, c, false, false);
}

// --- kernel 1: GT = [gamma * mobius_matvec(W,X) | gamma-1 | 1 | zeros]^T
//     stored TRANSPOSED: GT is (48 x N), GT[col*N + row]
__global__ void build_rhs_kernel(const float* __restrict__ Xin,
                                 const float* __restrict__ W,
                                 float* __restrict__ GT, int N) {
  int wave = threadIdx.x >> 5;
  int lane = threadIdx.x & 31;
  int row = blockIdx.x * 8 + wave;
  if (row >= N) return;
  float xv = Xin[row * DD + lane];
  // mx[lane] = sum_j X[row][j] * W[lane][j]   (X @ W^T)
  float mx = 0.f;
  #pragma unroll
  for (int j = 0; j < DD; ++j) {
    float xj = __shfl(xv, j, 32);
    mx = fmaf(xj, W[lane * DD + j], mx);
  }
  float xn  = fmaxf(sqrtf(wave_sum(xv * xv)), EPSF);
  float mxn = fmaxf(sqrtf(wave_sum(mx * mx)), EPSF);
  float scale = tanhf(mxn / xn * atanhf(clip1(xn))) / mxn;
  float m  = scale * mx;
  float m2 = wave_sum(m * m);
  float gamma = 2.0f / fmaxf(1.0f - m2, EPSF);   // lambda_x, K=-1
  GT[(size_t)lane * N + row] = gamma * m;
  if (lane < 16) {
    float v = 0.0f;
    if (lane == 0) v = gamma - 1.0f;  // midpoint denominator column
    else if (lane == 1) v = 1.0f;     // ones column -> alpha = rowsum(A)
    GT[(size_t)(32 + lane) * N + row] = v;
  }
}

// --- kernel 2: C(N x CT*16) = A(N x N) @ B, with B given transposed
//     BT is (CT*16 x N). fp32 WMMA 16x16x4, 2 row-tiles per wave,
//     8-way K-split per workgroup, LDS tree reduction.
//     All stream pointers are individual named scalars (not arrays) so the
//     address-space inference keeps every load a GLOBAL (not FLAT) vmem op.
template <int CT>
__global__ void __launch_bounds__(256) gemm_wmma_kernel(
    const float* __restrict__ A, const float* __restrict__ BT,
    float* __restrict__ C, int N) {
  const int LDC = CT * 16;
  int lane  = threadIdx.x & 31;
  int wavei = __builtin_amdgcn_readfirstlane(threadIdx.x >> 5);  // uniform SGPR
  int half  = lane >> 4;     // 0: lanes 0-15 (K,K+1)  1: lanes 16-31 (K+2,K+3)
  int m     = lane & 15;
  int row0  = blockIdx.x * 32;
  int Ks    = N >> 3;        // K-slice per wave (8 waves)
  int k0    = wavei * Ks;

  const float* pa0 = A + (size_t)(row0 + m) * N + 2 * half + k0;
  const float* pa1 = pa0 + (size_t)16 * N;
  const float* pb0 = BT + (size_t)m * N + 2 * half + k0;
  const float* pb1 = pb0 + (size_t)16 * N;   // used only when CT > 1
  const float* pb2 = pb0 + (size_t)32 * N;   // used only when CT > 2

  v8f a00 = {}, a01 = {}, a02 = {};
  v8f a10 = {}, a11 = {}, a12 = {};

  int chunks = Ks >> 4;                    // 16 K per chunk
  for (int it = 0; it < chunks; ++it) {
    __builtin_prefetch(pa0 + 256, 0, 3);   // ~1KB ahead on each A stream
    __builtin_prefetch(pa1 + 256, 0, 3);
    #pragma unroll
    for (int u = 0; u < 4; ++u) {          // constant offsets 0/16/32/48 B
      v2f A0 = *(const v2f*)(pa0 + 4 * u);
      v2f A1 = *(const v2f*)(pa1 + 4 * u);
      v2f B0 = *(const v2f*)(pb0 + 4 * u);
      a00 = wmma4(A0, B0, a00);
      a10 = wmma4(A1, B0, a10);
      if constexpr (CT > 1) {
        v2f B1 = *(const v2f*)(pb1 + 4 * u);
        a01 = wmma4(A0, B1, a01);
        a11 = wmma4(A1, B1, a11);
      }
      if constexpr (CT > 2) {
        v2f B2 = *(const v2f*)(pb2 + 4 * u);
        a02 = wmma4(A0, B2, a02);
        a12 = wmma4(A1, B2, a12);
      }
    }
    pa0 += 16; pa1 += 16; pb0 += 16;
    if constexpr (CT > 1) pb1 += 16;
    if constexpr (CT > 2) pb2 += 16;
  }

  __shared__ float red[8][2 * CT][8][32];
  #pragma unroll
  for (int j = 0; j < 8; ++j) {
    red[wavei][0][j][lane] = a00[j];
    red[wavei][CT][j][lane] = a10[j];
    if constexpr (CT > 1) {
      red[wavei][1][j][lane] = a01[j];
      red[wavei][CT + 1][j][lane] = a11[j];
    }
    if constexpr (CT > 2) {
      red[wavei][2][j][lane] = a02[j];
      red[wavei][CT + 2][j][lane] = a12[j];
    }
  }
  __syncthreads();
  if (wavei < 2 * CT) {
    int mt = wavei / CT;
    int t  = wavei % CT;
    #pragma unroll
    for (int j = 0; j < 8; ++j) {
      float s = 0.f;
      #pragma unroll
      for (int w = 0; w < 8; ++w) s += red[w][mt * CT + t][j][lane];
      int r = row0 + mt * 16 + j + 8 * half;   // C/D layout: VGPR j -> M=j / j+8
      C[(size_t)r * LDC + t * 16 + m] = s;
    }
  }
}

// --- kernel 3: midpoint -> mobius scalar mul -> sigma (logmap/relu/expmap)
__global__ void finish_layer_kernel(const float* __restrict__ OUT,
                                    float* __restrict__ H, int N) {
  int wave = threadIdx.x >> 5;
  int lane = threadIdx.x & 31;
  int row = blockIdx.x * 8 + wave;
  if (row >= N) return;
  float nom   = OUT[row * GCOLS + lane];
  float den   = OUT[row * GCOLS + 32];
  float alpha = OUT[row * GCOLS + 33];
  float s  = den >= 0.f ? 1.f : -1.f;
  float tm = nom / (s * fmaxf(fabsf(den), 1e-10f));       // two_mean
  float t2 = wave_sum(tm * tm);
  float am = tm / (1.0f + sqrtf(fmaxf(1.0f - t2, 0.0f))); // a_mean
  float an = fmaxf(sqrtf(wave_sum(am * am)), EPSF);
  float res = tanhf(alpha * atanhf(clip1(an))) * am / an; // mobius scalar mul
  float yn = fmaxf(sqrtf(wave_sum(res * res)), EPSF);
  float u  = atanhf(clip1(yn)) * res / yn;                // logmap0
  u = fmaxf(u, 0.0f);                                     // relu
  float un = fmaxf(sqrtf(wave_sum(u * u)), EPSF);
  H[row * DD + lane] = tanhf(un) * u / un;                // expmap0
}

// --- kernel 4: hyperbolic hyperplane logits -> LT stored TRANSPOSED (16 x N)
__global__ void logits_kernel(const float* __restrict__ H,
                              const float* __restrict__ Wl,   // D x C
                              const float* __restrict__ bks,  // C x D
                              float* __restrict__ LT, int N) {
  int gid = blockIdx.x * blockDim.x + threadIdx.x;
  if (gid >= N * CC) return;
  int n = gid >> 4;
  int k = gid & 15;
  float x2 = 0.f, y2 = 0.f, xy = 0.f, b2 = 0.f;
  #pragma unroll
  for (int d = 0; d < DD; ++d) {
    float b = bks[k * DD + d];
    float x = -b;
    float y = H[n * DD + d];
    x2 = fmaf(x, x, x2);
    y2 = fmaf(y, y, y2);
    xy = fmaf(x, y, xy);
    b2 = fmaf(b, b, b2);
  }
  // mobius_add(-b, y), K=-1: num=(1+2xy+y2)x + (1-x2)y ; den=1+2xy+x2*y2
  float c1 = 1.f + 2.f * xy + y2;
  float c2 = 1.f - x2;
  float den = fmaxf(1.f + 2.f * xy + x2 * y2, EPSF);
  float z2 = 0.f, za = 0.f, a2 = 0.f;
  #pragma unroll
  for (int d = 0; d < DD; ++d) {
    float b = bks[k * DD + d];
    float z = (c1 * (-b) + c2 * H[n * DD + d]) / den;
    float w = Wl[d * CC + k];
    z2 = fmaf(z, z, z2);
    za = fmaf(z, w, za);
    a2 = fmaf(w, w, a2);
  }
  float zn   = fmaxf(sqrtf(z2), 1e-10f);
  float anrm = fmaxf(sqrtf(a2), 1e-10f);
  float dist = asinhf(2.f * za / ((1.f - zn * zn) * anrm));
  float lam  = 2.f / fmaxf(1.f - b2, EPSF);
  LT[(size_t)k * N + n] = lam * anrm * dist;
}

extern "C" void kernel_launch(void* const* d_in, const int* in_sizes, int n_in,
                              void* d_out, int out_size, void* d_ws, size_t ws_size,
                              hipStream_t stream) {
  const float* X  = (const float*)d_in[0];
  const float* A  = (const float*)d_in[1];
  const float* W1 = (const float*)d_in[2];
  const float* W2 = (const float*)d_in[3];
  const float* Wl = (const float*)d_in[4];
  const float* pk = (const float*)d_in[5];
  float* out = (float*)d_out;
  int N = in_sizes[0] / DD;   // 8192

  float* GT  = (float*)d_ws;              // 48*N (transposed RHS)
  float* OUT = GT  + (size_t)GCOLS * N;   // N*48 (row-major)
  float* H   = OUT + (size_t)N * GCOLS;   // N*32
  float* LT  = H   + (size_t)N * DD;      // 16*N (transposed logits)

  dim3 blk(256);
  dim3 rows8(N / 8);
  dim3 tiles32(N / 32);

  // layer 1
  build_rhs_kernel<<<rows8, blk, 0, stream>>>(X, W1, GT, N);
  gemm_wmma_kernel<3><<<tiles32, blk, 0, stream>>>(A, GT, OUT, N);
  finish_layer_kernel<<<rows8, blk, 0, stream>>>(OUT, H, N);
  // layer 2
  build_rhs_kernel<<<rows8, blk, 0, stream>>>(H, W2, GT, N);
  gemm_wmma_kernel<3><<<tiles32, blk, 0, stream>>>(A, GT, OUT, N);
  finish_layer_kernel<<<rows8, blk, 0, stream>>>(OUT, H, N);
  // head + aggregation
  logits_kernel<<<(N * CC + 255) / 256, blk, 0, stream>>>(H, Wl, pk, LT, N);
  gemm_wmma_kernel<1><<<tiles32, blk, 0, stream>>>(A, LT, out, N);
}